// VQCNN_Base_12756052869708
// MI455X (gfx1250) — compile-verified
//
#include <hip/hip_runtime.h>

typedef float v2f __attribute__((ext_vector_type(2)));
typedef float v8f __attribute__((ext_vector_type(8)));

#define N_BATCH 16384

__device__ __forceinline__ float2 shflx2(float2 v, int mask) {
    float2 r;
    r.x = __shfl_xor(v.x, mask, 32);
    r.y = __shfl_xor(v.y, mask, 32);
    return r;
}

// wave-uniform broadcast from a compile-time lane -> v_readlane_b32 (SGPR source)
__device__ __forceinline__ float bcast(float v, int srclane) {
    return __uint_as_float(__builtin_amdgcn_readlane(__float_as_uint(v), srclane));
}

__device__ __forceinline__ float fast_rcp(float x) {
    return __builtin_amdgcn_rcpf(x);           // v_rcp_f32
}

__device__ __forceinline__ float fast_sigmoid(float z) {
    return fast_rcp(1.0f + __expf(-z));        // v_exp_f32 + v_rcp_f32
}

__device__ __forceinline__ float fast_tanh(float z) {
    // tanh(z) = 1 - 2/(1 + e^{2z}); branch-free, saturates correctly at +-1
    return 1.0f - 2.0f * fast_rcp(1.0f + __expf(2.0f * z));
}

// One batch element per wave32. Amplitude flat index i = lane*8 + k.
// Qubit q <-> bit (7-q) of i. Bits 0..2 = register bits (k), bits 3..7 = lane bits.
__global__ __launch_bounds__(256) void vqc_state_kernel(
    const float* __restrict__ x,    // [B,4,8]
    const float* __restrict__ crx,  // [10]
    float* __restrict__ feats)      // [B,2]
{
    const int lane = threadIdx.x & 31;
    const int wave = threadIdx.x >> 5;
    const int b = blockIdx.x * 8 + wave;

    float2 a[8];
#pragma unroll
    for (int k = 0; k < 8; ++k) { a[k].x = 0.0f; a[k].y = 0.0f; }
    a[0].x = (lane == 0) ? 1.0f : 0.0f;   // |0...0>

    // coalesced load of this element's 32 angles, one per lane; each lane
    // computes cos/sin of its own half-angle ONCE (2 trans ops total).
    const float xl = x[(size_t)b * 32 + lane];
    const float cl = __cosf(0.5f * xl);
    const float sl = __sinf(0.5f * xl);

    // Fused lane permutation for the CNOT chain q=0..3 (all lane-bit gates):
    // new[L] = old[srcLane(L)], srcLane = (L&16) | ((L ^ (L>>1)) & 15)
    const int ringsrc = (lane & 16) | ((lane ^ (lane >> 1)) & 15);

#pragma unroll
    for (int cyc = 0; cyc < 4; ++cyc) {
        // ---- RY layer: q = 0..7 ----
#pragma unroll
        for (int q = 0; q < 8; ++q) {
            const float c = bcast(cl, cyc * 8 + q);
            const float s = bcast(sl, cyc * 8 + q);
            const int p = 7 - q;                  // bit position of qubit q
            if (p >= 3) {                         // cross-lane pair
                const int lb = p - 3;
                const float sgn = ((lane >> lb) & 1) ? s : -s;
#pragma unroll
                for (int k = 0; k < 8; ++k) {
                    float2 o = shflx2(a[k], 1 << lb);
                    a[k].x = c * a[k].x + sgn * o.x;
                    a[k].y = c * a[k].y + sgn * o.y;
                }
            } else {                              // register-local pair
                const int m = 1 << p;
#pragma unroll
                for (int k = 0; k < 8; ++k) {
                    if ((k & m) == 0) {
                        const int k1 = k | m;
                        float2 a0 = a[k], a1 = a[k1];
                        a[k].x  = c * a0.x - s * a1.x;  a[k].y  = c * a0.y - s * a1.y;
                        a[k1].x = s * a0.x + c * a1.x;  a[k1].y = s * a0.y + c * a1.y;
                    }
                }
            }
        }
        // ---- CNOT ring ----
        // q=0..3 fused: single lane-gather per register
#pragma unroll
        for (int k = 0; k < 8; ++k) {
            a[k].x = __shfl(a[k].x, ringsrc, 32);
            a[k].y = __shfl(a[k].y, ringsrc, 32);
        }
        // q=4: control lane bit0, target register bit2 -> conditional reg swap
        {
            const bool ctl = lane & 1;
#pragma unroll
            for (int k = 0; k < 4; ++k) {
                float2 t0 = a[k], t1 = a[k + 4];
                a[k].x     = ctl ? t1.x : t0.x;  a[k].y     = ctl ? t1.y : t0.y;
                a[k + 4].x = ctl ? t0.x : t1.x;  a[k + 4].y = ctl ? t0.y : t1.y;
            }
        }
        // q=5: control k bit2, target k bit1 -> swap (4,6),(5,7)  (free renames)
        { float2 t = a[4]; a[4] = a[6]; a[6] = t; t = a[5]; a[5] = a[7]; a[7] = t; }
        // q=6: control k bit1, target k bit0 -> swap (2,3),(6,7)
        { float2 t = a[2]; a[2] = a[3]; a[3] = t; t = a[6]; a[6] = a[7]; a[7] = t; }
        // q=7: control k bit0 (odd k), target lane bit4
        {
#pragma unroll
            for (int k = 1; k < 8; k += 2) a[k] = shflx2(a[k], 16);
        }
    }

    // ---- CRX tail: gate i on (c = i%8, t = c+1 mod 8); angles are scalar loads ----
#pragma unroll
    for (int i = 0; i < 10; ++i) {
        const float th = 0.5f * crx[i];          // uniform -> s_load
        const float co = __cosf(th), s = __sinf(th);
        const int c = i & 7;
        if (c <= 3) {                            // control lane bit (4-c), target lane bit (3-c)
            const int cb = 4 - c, tb = 3 - c;
            const bool ctl = (lane >> cb) & 1;
#pragma unroll
            for (int k = 0; k < 8; ++k) {
                float2 o = shflx2(a[k], 1 << tb);
                const float nr = co * a[k].x + s * o.y;
                const float ni = co * a[k].y - s * o.x;
                a[k].x = ctl ? nr : a[k].x;
                a[k].y = ctl ? ni : a[k].y;
            }
        } else if (c == 4) {                     // control lane bit0, target k bit2
            const bool ctl = lane & 1;
            float2 n[8];
#pragma unroll
            for (int k = 0; k < 8; ++k) {
                float2 o = a[k ^ 4];
                n[k].x = co * a[k].x + s * o.y;
                n[k].y = co * a[k].y - s * o.x;
            }
#pragma unroll
            for (int k = 0; k < 8; ++k) {
                a[k].x = ctl ? n[k].x : a[k].x;
                a[k].y = ctl ? n[k].y : a[k].y;
            }
        } else if (c == 5) {                     // control k bit2, target k bit1: pairs (4,6),(5,7)
            float2 n[4];
#pragma unroll
            for (int k = 4; k < 8; ++k) {
                float2 o = a[k ^ 2];
                n[k - 4].x = co * a[k].x + s * o.y;
                n[k - 4].y = co * a[k].y - s * o.x;
            }
#pragma unroll
            for (int k = 4; k < 8; ++k) a[k] = n[k - 4];
        } else if (c == 6) {                     // control k bit1, target k bit0: pairs (2,3),(6,7)
            float2 n2, n3, n6, n7;
            n2.x = co * a[2].x + s * a[3].y;  n2.y = co * a[2].y - s * a[3].x;
            n3.x = co * a[3].x + s * a[2].y;  n3.y = co * a[3].y - s * a[2].x;
            n6.x = co * a[6].x + s * a[7].y;  n6.y = co * a[6].y - s * a[7].x;
            n7.x = co * a[7].x + s * a[6].y;  n7.y = co * a[7].y - s * a[6].x;
            a[2] = n2; a[3] = n3; a[6] = n6; a[7] = n7;
        } else {                                 // c==7: control k bit0 (odd k), target lane bit4
#pragma unroll
            for (int k = 1; k < 8; k += 2) {
                float2 o = shflx2(a[k], 16);
                const float nr = co * a[k].x + s * o.y;
                const float ni = co * a[k].y - s * o.x;
                a[k].x = nr; a[k].y = ni;
            }
        }
    }

    // ---- measurement: <Z> on qubit 3 (bit4 -> lane bit1) and qubit 7 (bit0 -> k bit0) ----
    float sum_all = 0.0f, sum_k0 = 0.0f;
#pragma unroll
    for (int k = 0; k < 8; ++k) {
        const float pk = a[k].x * a[k].x + a[k].y * a[k].y;
        sum_all += pk;
        sum_k0 += (k & 1) ? -pk : pk;
    }
    float f0 = ((lane >> 1) & 1) ? -sum_all : sum_all;  // qubit 3
    float f1 = sum_k0;                                  // qubit 7
#pragma unroll
    for (int m = 16; m >= 1; m >>= 1) {
        f0 += __shfl_xor(f0, m, 32);
        f1 += __shfl_xor(f1, m, 32);
    }
    if (lane == 0) {
        float2 fo; fo.x = f0; fo.y = f1;
        *(float2*)(feats + 2 * b) = fo;
    }
}

// MLP head via WMMA f32 16x16x4: H = tanh(F @ W1^T + b1), out = sigmoid(H @ w2^T + b2).
// One wave handles 16 batch rows. A: 16x4 (K=2 real, K=2..3 zero-padded),
// B: 4x16 = W1^T zero-padded, C: bias broadcast per column. The w2 dot is a
// 16-lane shfl_xor reduction (no LDS, no barrier).
__global__ __launch_bounds__(256) void vqc_head_kernel(
    const float* __restrict__ feats, const float* __restrict__ w1,
    const float* __restrict__ b1, const float* __restrict__ w2,
    const float* __restrict__ b2, float* __restrict__ out)
{
    const int lane = threadIdx.x & 31;
    const int wave = threadIdx.x >> 5;
    const int half = lane >> 4;          // 0: lanes 0-15 (K=0,1 / rows 0-7), 1: lanes 16-31
    const int n = lane & 15;
    const int rowbase = (blockIdx.x * 8 + wave) * 16;

    // A operand: lane<16 -> row n, K=0/1 in v0/v1; lanes>=16 -> K=2/3 = 0
    const int r = rowbase + n;           // always in range
    const float fa = feats[2 * r];
    const float fb = feats[2 * r + 1];
    v2f av; av.x = half ? 0.0f : fa;  av.y = half ? 0.0f : fb;

    // B operand: VGPR j, lanes<16 -> K=j, N=n; lanes>=16 -> K=j+2 (zero pad)
    const int nn = (n < 10) ? n : 0;
    const float wv0 = w1[nn * 2];
    const float wv1 = w1[nn * 2 + 1];
    const bool bvalid = (half == 0) && (n < 10);
    v2f bv; bv.x = bvalid ? wv0 : 0.0f;  bv.y = bvalid ? wv1 : 0.0f;

    // C operand: bias per column, replicated over the 8 row-VGPRs
    const float bias = (n < 10) ? b1[nn] : 0.0f;
    v8f cv;
#pragma unroll
    for (int j = 0; j < 8; ++j) cv[j] = bias;

    v8f d = __builtin_amdgcn_wmma_f32_16x16x4_f32(
        /*neg_a=*/false, av, /*neg_b=*/false, bv,
        /*c_mod=*/(short)0, cv, /*reuse_a=*/false, /*reuse_b=*/false);

    // D layout: VGPR j -> row (half*8 + j), col n.
    // Per row: sum_n w2[n]*tanh(h).  Branch-free tanh via v_exp_f32 + v_rcp_f32.
    const float w2n = (n < 10) ? w2[nn] : 0.0f;
    float rs[8];
#pragma unroll
    for (int j = 0; j < 8; ++j) rs[j] = w2n * fast_tanh(d[j]);
#pragma unroll
    for (int m = 1; m <= 8; m <<= 1) {
#pragma unroll
        for (int j = 0; j < 8; ++j) rs[j] += __shfl_xor(rs[j], m, 32);
    }

    if ((lane & 15) == 0) {              // lane 0 -> rows 0-7, lane 16 -> rows 8-15
        const float bb = b2[0];
        float4 o0, o1;
        o0.x = fast_sigmoid(rs[0] + bb);  o0.y = fast_sigmoid(rs[1] + bb);
        o0.z = fast_sigmoid(rs[2] + bb);  o0.w = fast_sigmoid(rs[3] + bb);
        o1.x = fast_sigmoid(rs[4] + bb);  o1.y = fast_sigmoid(rs[5] + bb);
        o1.z = fast_sigmoid(rs[6] + bb);  o1.w = fast_sigmoid(rs[7] + bb);
        float4* op = (float4*)(out + rowbase + half * 8);
        op[0] = o0;
        op[1] = o1;
    }
}

extern "C" void kernel_launch(void* const* d_in, const int* in_sizes, int n_in,
                              void* d_out, int out_size, void* d_ws, size_t ws_size,
                              hipStream_t stream) {
    const float* x   = (const float*)d_in[0];  // [16384,4,8]
    const float* crx = (const float*)d_in[1];  // [10]
    const float* w1  = (const float*)d_in[2];  // [10,2]
    const float* b1  = (const float*)d_in[3];  // [10]
    const float* w2  = (const float*)d_in[4];  // [1,10]
    const float* b2  = (const float*)d_in[5];  // [1]

    float* feats = (float*)d_ws;               // [16384,2] scratch

    // 8 waves (one batch element each) per 256-thread block
    vqc_state_kernel<<<N_BATCH / 8, 256, 0, stream>>>(x, crx, feats);
    // 16 rows per wave, 8 waves per block -> 128 rows/block
    vqc_head_kernel<<<N_BATCH / 128, 256, 0, stream>>>(feats, w1, b1, w2, b2,
                                                       (float*)d_out);
}